// BayesianOddLayer_52793738002720
// MI455X (gfx1250) — compile-verified
//
#include <hip/hip_runtime.h>
#include <stdint.h>

#define N_VARS   1024
#define N_CHECKS 512
#define VAR_DEG  3
#define N_EDGES  (N_VARS * VAR_DEG)      // 3072
#define BATCH    2048
#define CLIP_TANH 10.0f
#define E_TILE   768                     // 3 edges per thread, divisible by 3
#define V_TILE   (E_TILE / VAR_DEG)      // 256 variables per chunk
#define HALF_N   ((BATCH / 2) * N_EDGES) // 3145728: jax uniform counter split point

// ---------------------------------------------------------------------------
// Bit-exact JAX threefry2x32 (20 rounds), usable on host (key derivation for
// jax.random.split) and device (per-element uniforms).
// ---------------------------------------------------------------------------
__host__ __device__ static inline uint32_t rotl32(uint32_t x, int r) {
    return (x << r) | (x >> (32 - r));
}

__host__ __device__ static inline void threefry2x32(
    uint32_t k0, uint32_t k1, uint32_t x0, uint32_t x1,
    uint32_t& o0, uint32_t& o1)
{
    const uint32_t ks2 = k0 ^ k1 ^ 0x1BD11BDAu;
    x0 += k0; x1 += k1;
    // group 1: rotations {13,15,26,6}
    x0 += x1; x1 = rotl32(x1, 13); x1 ^= x0;
    x0 += x1; x1 = rotl32(x1, 15); x1 ^= x0;
    x0 += x1; x1 = rotl32(x1, 26); x1 ^= x0;
    x0 += x1; x1 = rotl32(x1,  6); x1 ^= x0;
    x0 += k1;  x1 += ks2 + 1u;
    // group 2: rotations {17,29,16,24}
    x0 += x1; x1 = rotl32(x1, 17); x1 ^= x0;
    x0 += x1; x1 = rotl32(x1, 29); x1 ^= x0;
    x0 += x1; x1 = rotl32(x1, 16); x1 ^= x0;
    x0 += x1; x1 = rotl32(x1, 24); x1 ^= x0;
    x0 += ks2; x1 += k0 + 2u;
    // group 3
    x0 += x1; x1 = rotl32(x1, 13); x1 ^= x0;
    x0 += x1; x1 = rotl32(x1, 15); x1 ^= x0;
    x0 += x1; x1 = rotl32(x1, 26); x1 ^= x0;
    x0 += x1; x1 = rotl32(x1,  6); x1 ^= x0;
    x0 += k0;  x1 += k1 + 3u;
    // group 4
    x0 += x1; x1 = rotl32(x1, 17); x1 ^= x0;
    x0 += x1; x1 = rotl32(x1, 29); x1 ^= x0;
    x0 += x1; x1 = rotl32(x1, 16); x1 ^= x0;
    x0 += x1; x1 = rotl32(x1, 24); x1 ^= x0;
    x0 += k1;  x1 += ks2 + 4u;
    // group 5
    x0 += x1; x1 = rotl32(x1, 13); x1 ^= x0;
    x0 += x1; x1 = rotl32(x1, 15); x1 ^= x0;
    x0 += x1; x1 = rotl32(x1, 26); x1 ^= x0;
    x0 += x1; x1 = rotl32(x1,  6); x1 ^= x0;
    x0 += ks2; x1 += k0 + 5u;
    o0 = x0; o1 = x1;
}

__device__ static inline float u01(uint32_t bits) {
    // jax: bitcast((bits >> 9) | 0x3f800000, f32) - 1.0
    return __uint_as_float((bits >> 9) | 0x3f800000u) - 1.0f;
}

__device__ static inline float tanh_half(float o) {
    // tanh(0.5*o) = (e^o - 1) * rcp(e^o + 1); |o| <= 10 so e^o is safe
    const float t = __expf(o);
    return (t - 1.0f) * __builtin_amdgcn_rcpf(t + 1.0f);
}

// CDNA5 async-tensor data movement: global -> LDS without touching VGPRs.
__device__ static inline void async_load_b128(unsigned lds_off, const float* gptr) {
    asm volatile("global_load_async_to_lds_b128 %0, %1, off"
                 :: "v"(lds_off), "v"((unsigned long long)(uintptr_t)gptr)
                 : "memory");
}
__device__ static inline void wait_async_zero() {
    asm volatile("s_wait_asynccnt 0x0" ::: "memory");
}

// ---------------------------------------------------------------------------
// Kernel 1: fold the Tanner-graph masks into compact per-edge tables.
// Per edge e: c[k] = masked odd_weights[3*(e/3)+k, e] (k=0..2), skip weight,
// keep_prob. 60 KB total -> L2-resident, reused by all 2048 batch rows.
// ---------------------------------------------------------------------------
__global__ void bp_odd_precompute(
    const float* __restrict__ odd_w, const float* __restrict__ llr_w,
    const float* __restrict__ logits, const int* __restrict__ edge_var,
    const int* __restrict__ edge_chk, float4* __restrict__ c4,
    float* __restrict__ kparr)
{
    const int e = blockIdx.x * blockDim.x + threadIdx.x;
    if (e >= N_EDGES) return;
    const int v    = edge_var[e];
    const int chk  = edge_chk[e];
    const int base = (e / VAR_DEG) * VAR_DEG;   // var-major grouping of edges
    float c[VAR_DEG];
#pragma unroll
    for (int k = 0; k < VAR_DEG; ++k) {
        const int ek = base + k;
        const bool m = (edge_var[ek] == v) && (edge_chk[ek] != chk); // extrinsic mask
        c[k] = m ? odd_w[(size_t)ek * N_EDGES + e] : 0.0f;
    }
    c4[e]    = make_float4(c[0], c[1], c[2], llr_w[(size_t)v * N_EDGES + e]);
    kparr[e] = 1.0f / (1.0f + __expf(-logits[e]));  // sigmoid
}

// ---------------------------------------------------------------------------
// Kernel 2: fused sparse-matmul + threefry dropout + tanh ensemble.
// Block = (b, b+1024) batch-row pair x one 768-edge chunk; x and llr slices
// are staged in LDS with async-tensor loads; every threefry eval serves both
// rows (JAX counter layout pairs idx and idx + n/2 = idx + 1024*E).
// ---------------------------------------------------------------------------
__global__ __launch_bounds__(256) void bp_odd_main(
    const float* __restrict__ x, const float* __restrict__ llr,
    const float4* __restrict__ c4, const float* __restrict__ kparr,
    float* __restrict__ out,
    uint32_t k00, uint32_t k01, uint32_t k10, uint32_t k11,
    uint32_t k20, uint32_t k21)
{
    __shared__ __align__(16) float xs[2][E_TILE];   // 6 KB
    __shared__ __align__(16) float ls[2][V_TILE];   // 2 KB
    const int t  = threadIdx.x;
    const int b  = blockIdx.y;            // 0..1023 (low half of the batch)
    const int bh = b + BATCH / 2;         // paired high-half row
    const int e0 = blockIdx.x * E_TILE;
    const int v0 = e0 / VAR_DEG;

    if (t < E_TILE / 4) {                 // 192 lanes x 16B = 3 KB per x slice
        async_load_b128((unsigned)(uintptr_t)&xs[0][t * 4],
                        x + (size_t)b  * N_EDGES + e0 + t * 4);
        async_load_b128((unsigned)(uintptr_t)&xs[1][t * 4],
                        x + (size_t)bh * N_EDGES + e0 + t * 4);
    }
    if (t < V_TILE / 4) {                 // 64 lanes x 16B = 1 KB per llr slice
        async_load_b128((unsigned)(uintptr_t)&ls[0][t * 4],
                        llr + (size_t)b  * N_VARS + v0 + t * 4);
        async_load_b128((unsigned)(uintptr_t)&ls[1][t * 4],
                        llr + (size_t)bh * N_VARS + v0 + t * 4);
    }
    wait_async_zero();
    __syncthreads();

    const float inv3 = 1.0f / 3.0f;
#pragma unroll
    for (int j = 0; j < E_TILE / 256; ++j) {
        const int el = j * 256 + t;
        const int e  = e0 + el;
        const int lv = el / VAR_DEG;                 // local variable index
        const int lb = lv * VAR_DEG;                 // group base within tile
        const float4 c  = c4[e];
        const float kpe = kparr[e];

        // x @ (mask*odd_weights): 2 nonzeros (+1 self term zeroed by mask)
        const float xaL = c.x * xs[0][lb] + c.y * xs[0][lb + 1] + c.z * xs[0][lb + 2];
        const float xaH = c.x * xs[1][lb] + c.y * xs[1][lb + 1] + c.z * xs[1][lb + 2];
        // llr @ (mask*llr_weights): 1 nonzero
        const float owlL = ls[0][lv] * c.w;
        const float owlH = ls[1][lv] * c.w;

        const uint32_t idxL = (uint32_t)b * N_EDGES + (uint32_t)e;
        const uint32_t idxH = idxL + (uint32_t)HALF_N;

        float sL = 0.0f, sH = 0.0f;
        uint32_t r0, r1;
        // ensemble member 0
        threefry2x32(k00, k01, idxL, idxH, r0, r1);
        { float o = fminf(fmaxf(xaL + (u01(r0) < kpe ? owlL : 0.0f), -CLIP_TANH), CLIP_TANH); sL += tanh_half(o); }
        { float o = fminf(fmaxf(xaH + (u01(r1) < kpe ? owlH : 0.0f), -CLIP_TANH), CLIP_TANH); sH += tanh_half(o); }
        // ensemble member 1
        threefry2x32(k10, k11, idxL, idxH, r0, r1);
        { float o = fminf(fmaxf(xaL + (u01(r0) < kpe ? owlL : 0.0f), -CLIP_TANH), CLIP_TANH); sL += tanh_half(o); }
        { float o = fminf(fmaxf(xaH + (u01(r1) < kpe ? owlH : 0.0f), -CLIP_TANH), CLIP_TANH); sH += tanh_half(o); }
        // ensemble member 2
        threefry2x32(k20, k21, idxL, idxH, r0, r1);
        { float o = fminf(fmaxf(xaL + (u01(r0) < kpe ? owlL : 0.0f), -CLIP_TANH), CLIP_TANH); sL += tanh_half(o); }
        { float o = fminf(fmaxf(xaH + (u01(r1) < kpe ? owlH : 0.0f), -CLIP_TANH), CLIP_TANH); sH += tanh_half(o); }

        out[(size_t)idxL] = sL * inv3;
        out[(size_t)idxH] = sH * inv3;
    }
}

// ---------------------------------------------------------------------------
extern "C" void kernel_launch(void* const* d_in, const int* in_sizes, int n_in,
                              void* d_out, int out_size, void* d_ws, size_t ws_size,
                              hipStream_t stream) {
    (void)in_sizes; (void)n_in; (void)out_size; (void)ws_size;
    const float* x        = (const float*)d_in[0];
    const float* llr      = (const float*)d_in[1];
    const float* odd_w    = (const float*)d_in[2];
    const float* llr_w    = (const float*)d_in[3];
    const float* logits   = (const float*)d_in[4];
    const int*   edge_var = (const int*)d_in[5];
    const int*   edge_chk = (const int*)d_in[6];
    float*       out      = (float*)d_out;

    char* ws = (char*)d_ws;
    float4* c4    = (float4*)ws;                                     // 48 KB
    float*  kparr = (float*)(ws + (size_t)N_EDGES * sizeof(float4)); // 12 KB

    // Replicate jax.random.split(jax.random.key(42), 3) on the host:
    // key = (0, 42); counts = iota(6) -> threefry over (i, 3+i); keys come from
    // concat(out0, out1).reshape(3, 2).
    uint32_t f0[3], f1[3];
    for (int i = 0; i < 3; ++i)
        threefry2x32(0u, 42u, (uint32_t)i, (uint32_t)(3 + i), f0[i], f1[i]);
    const uint32_t K[6] = { f0[0], f0[1], f0[2], f1[0], f1[1], f1[2] };

    bp_odd_precompute<<<(N_EDGES + 255) / 256, 256, 0, stream>>>(
        odd_w, llr_w, logits, edge_var, edge_chk, c4, kparr);

    dim3 grid(N_EDGES / E_TILE, BATCH / 2);   // 4 x 1024 blocks
    bp_odd_main<<<grid, 256, 0, stream>>>(
        x, llr, c4, kparr, out,
        K[0], K[1], K[2], K[3], K[4], K[5]);
}